// BatchBlurk_82429012345594
// MI455X (gfx1250) — compile-verified
//
#include <hip/hip_runtime.h>
#include <hip/hip_bf16.h>

// ---------------------------------------------------------------------------
// BatchBlurk: per-batch 21x21 depthwise blur (jnp reflect pad) + MATLAB-style
// antialiased bicubic downsample x4 (symmetric boundary).
//
// Optimized pipeline (column filter folded into the blur -> 2.3x fewer FLOPs):
//   prep_gtab : per-batch fused 21x40 stride-4 column kernels (5 classes)
//   prep_avar : banded 16x80 row-downsample A-matrix (3 variants)
//   convdsw   : fused blur+col-downsample -> z (24,1024,256); interior tiles
//               fetched via Tensor Data Mover (tensor_load_to_lds, 6-arg form)
//   dsrow     : row downsample as WMMA f32 16x16x4 banded GEMM -> out
// ---------------------------------------------------------------------------

typedef __attribute__((ext_vector_type(2))) float v2f;
typedef __attribute__((ext_vector_type(8))) float v8f;
typedef __attribute__((ext_vector_type(4))) unsigned int u32x4;
typedef __attribute__((ext_vector_type(8))) int i32x8;
typedef __attribute__((ext_vector_type(4))) int i32x4;

#define H 1024
#define W 1024
#define BC 24
#define OUTD 256
#define AW 80           // banded A block K extent
#define GW 40           // fused column-kernel width (38 + pad)
#define GT (21 * GW)    // floats per class table
#define PW 100          // LDS patch row stride (floats, 16B-aligned rows)
#define PR 52           // patch rows (32 outputs + 20 halo)

#if defined(__has_builtin)
#if __has_builtin(__builtin_amdgcn_tensor_load_to_lds) && \
    __has_builtin(__builtin_amdgcn_s_wait_tensorcnt)
#define USE_TDM 1
#endif
#endif

__device__ __forceinline__ float bicubic_w(int j) {
    // 0.25 * cubic((8.5 - j)/4), MATLAB a=-0.5 kernel (un-normalized)
    float ax = fabsf((8.5f - (float)j) * 0.25f);
    float wv;
    if (ax <= 1.f)      wv = (1.5f * ax - 2.5f) * ax * ax + 1.f;
    else if (ax <= 2.f) wv = ((-0.5f * ax + 2.5f) * ax - 4.f) * ax + 2.f;
    else                wv = 0.f;
    return wv * 0.25f;
}

// ---------------------------------------------------------------------------
// gtab[b][t][ky][b40]: fused column kernel.  Classes t: 0->p=0, 1->p=1,
// 2->generic, 3->p=254, 4->p=255.  g = (mirror-folded 18-tap) conv k-row.
// Bases chosen 16B-aligned vs patch: generic x0=4p-17; t0/t1 x0=-13 (shift 3);
// t3 x0=999; t4 x0=1003.
// ---------------------------------------------------------------------------
__global__ __launch_bounds__(256) void prep_gtab(const float* __restrict__ kern,
                                                 float* __restrict__ gtab) {
    float ww[18];
    float s = 0.f;
    #pragma unroll
    for (int j = 0; j < 18; ++j) { ww[j] = bicubic_w(j); s += ww[j]; }
    float inv = 1.f / s;
    #pragma unroll
    for (int j = 0; j < 18; ++j) ww[j] *= inv;

    const int total = 8 * 5 * GT;
    for (int e = blockIdx.x * blockDim.x + threadIdx.x; e < total;
         e += gridDim.x * blockDim.x) {
        int b  = e / (5 * GT);
        int r  = e % (5 * GT);
        int t  = r / GT;
        int r2 = r % GT;
        int ky = r2 / GW;
        int bb = r2 % GW;

        float wf[18];
        #pragma unroll
        for (int c = 0; c < 18; ++c) wf[c] = 0.f;
        int shift = 0;
        if (t == 2) {
            #pragma unroll
            for (int i = 0; i < 18; ++i) wf[i] = ww[i];
        } else {
            int ph = (t == 0) ? 0 : (t == 1) ? 1 : (t == 3) ? 254 : 255;
            int cb = (t <= 1) ? 0 : (t == 3) ? 1009 : 1013;
            shift  = (t <= 1) ? 3 : 0;
            for (int i = 0; i < 18; ++i) {
                int idx = 4 * ph - 7 + i;                 // symmetric mirror
                int m = (idx < 0) ? (-idx - 1) : ((idx > 1023) ? (2047 - idx) : idx);
                wf[m - cb] += ww[i];
            }
        }
        float acc = 0.f;
        const float* kb = kern + b * 441 + ky * 21;
        for (int c = 0; c < 18; ++c) {
            int kx = bb - shift - c;
            if (kx >= 0 && kx < 21) acc += wf[c] * kb[kx];
        }
        gtab[e] = acc;
    }
}

// ---------------------------------------------------------------------------
// Avar (3 x 16 x 80): banded row-resize block. v0 first tile (base row 0),
// v1 interior (base 4*o0-7), v2 last tile (base 953). Symmetric mirror folds.
// ---------------------------------------------------------------------------
__global__ __launch_bounds__(256) void prep_avar(float* __restrict__ Avar) {
    float ww[18];
    float s = 0.f;
    #pragma unroll
    for (int j = 0; j < 18; ++j) { ww[j] = bicubic_w(j); s += ww[j]; }
    float inv = 1.f / s;
    #pragma unroll
    for (int j = 0; j < 18; ++j) ww[j] *= inv;

    for (int e = threadIdx.x; e < 3 * 16 * AW; e += blockDim.x) {
        int v = e / (16 * AW);
        int r = e % (16 * AW);
        int m = r / AW;
        int l = r % AW;
        float acc = 0.f;
        #pragma unroll
        for (int jj = 0; jj < 18; ++jj) {
            int local;
            if (v == 0) {
                int idx = 4 * m - 7 + jj;
                local = (idx < 0) ? (-idx - 1) : idx;
            } else if (v == 1) {
                local = 4 * m + jj;
            } else {
                int idx = 4 * (240 + m) - 7 + jj;
                int mi = (idx >= 1024) ? (2047 - idx) : idx;
                local = mi - 953;
            }
            if (local == l) acc += ww[jj];
        }
        Avar[e] = acc;
    }
}

// ---------------------------------------------------------------------------
// convdsw: z[bc,h,p] = sum_ky sum_b g[class(p)][ky][b] * x_reflect[h+ky-10,
//                                                     xbase(p)+b]
// Block = 16p x 32h tile.  Patch 52x100 in LDS.  Interior tiles via TDM.
// ---------------------------------------------------------------------------
__global__ __launch_bounds__(256) void convdsw_kernel(const float* __restrict__ x,
                                                      const float* __restrict__ gtab,
                                                      float* __restrict__ z) {
    __shared__ float sp[PR * PW];      // 20.8 KB patch
    __shared__ float sg[3 * GT];       // 10.1 KB coeffs: [gen, edgeA, edgeB]

    const int bc  = blockIdx.z;
    const int b   = bc / 3;
    const int pt  = blockIdx.x;        // 16 p-tiles
    const int ht  = blockIdx.y;        // 32 h-tiles
    const int p0  = pt * 16;
    const int h0  = ht * 32;
    const int cb  = 4 * p0 - 17;       // patch x-column base
    const int tid = threadIdx.x;

    const float* gb = gtab + b * (5 * GT);
    for (int i = tid; i < GT; i += 256) {
        sg[i] = gb[2 * GT + i];                        // generic
        if (pt == 0)       { sg[GT + i] = gb[i];          sg[2 * GT + i] = gb[GT + i]; }
        else if (pt == 15) { sg[GT + i] = gb[3 * GT + i]; sg[2 * GT + i] = gb[4 * GT + i]; }
    }

    const float* xb = x + (size_t)bc * (H * W);
    const bool interior = (pt >= 1) && (pt <= 14) && (ht >= 1) && (ht <= 30);
    bool need_valu = true;

#ifdef USE_TDM
    if (interior) {
        if (tid < 32) {
            // D# group0: count=1, lds_addr, 57b global addr, type=2
            unsigned long long ga =
                (unsigned long long)(size_t)(xb + (size_t)(h0 - 10) * W + cb);
            u32x4 g0;
            g0[0] = 1u;
            g0[1] = (unsigned int)(size_t)sp;
            g0[2] = (unsigned int)(ga & 0xffffffffull);
            g0[3] = ((unsigned int)(ga >> 32) & 0x01ffffffu) | (2u << 30);
            // D# group1: data_size=4B, tensor 1024x1024 (stride 1024, 1M),
            // tile 100x52, no pad/iterate/multicast
            i32x8 g1;
            g1[0] = (int)(2u << 16);            // data_size code 2 (4 bytes)
            g1[1] = (int)(1024u << 16);         // tensor_dim0 lo16
            g1[2] = (int)(1024u << 16);         // dim0 hi16=0 | tensor_dim1 lo16
            g1[3] = (int)(100u << 16);          // dim1 hi16=0 | tile_dim0=100
            g1[4] = (int)52;                    // tile_dim1=52, tile_dim2=0
            g1[5] = (int)1024;                  // tensor_dim0_stride lo32
            g1[6] = 0;                          // stride0 hi | stride1 lo16 (=0)
            g1[7] = (int)((1024u * 1024u) >> 16); // tensor_dim1_stride[47:16]
            i32x4 g2; g2[0] = 0; g2[1] = 0; g2[2] = 0; g2[3] = 0;
            i32x4 g3; g3[0] = 0; g3[1] = 0; g3[2] = 0; g3[3] = 0;
            i32x8 g4;
            #pragma unroll
            for (int q = 0; q < 8; ++q) g4[q] = 0;
            // amdgpu-toolchain (clang-23) 6-arg form:
            // (u32x4, i32x8, i32x4, i32x4, i32x8, i32 cpol)
            __builtin_amdgcn_tensor_load_to_lds(g0, g1, g2, g3, g4, 0);
            __builtin_amdgcn_s_wait_tensorcnt(0);
        }
        need_valu = false;
    }
#endif
    if (need_valu) {
        for (int i = tid; i < PR * PW; i += 256) {
            int pr = i / PW, pc = i % PW;
            int gr = h0 - 10 + pr;                     // jnp 'reflect': -i -> i
            gr = (gr < 0) ? -gr : ((gr > H - 1) ? (2 * H - 2 - gr) : gr);
            int gc = cb + pc;
            gc = (gc < 0) ? -gc : ((gc > W - 1) ? (2 * W - 2 - gc) : gc);
            sp[i] = xb[gr * W + gc];
        }
    }
    __syncthreads();

    const int tp = tid & 15;   // p within tile
    const int hg = tid >> 4;   // h group (2 rows each)

    const float* ct;
    int pc0;
    if (pt == 0 && tp == 0)        { ct = sg + GT;     pc0 = 4;  }
    else if (pt == 0 && tp == 1)   { ct = sg + 2 * GT; pc0 = 4;  }
    else if (pt == 15 && tp == 14) { ct = sg + GT;     pc0 = 56; }
    else if (pt == 15 && tp == 15) { ct = sg + 2 * GT; pc0 = 60; }
    else                           { ct = sg;          pc0 = 4 * tp; }

    float acc0 = 0.f, acc1 = 0.f;
    for (int ky = 0; ky < 21; ++ky) {
        const float4* g4 = (const float4*)(ct + ky * GW);
        const float4* r0 = (const float4*)(sp + (hg + ky) * PW + pc0);
        const float4* r1 = (const float4*)(sp + (hg + 16 + ky) * PW + pc0);
        #pragma unroll
        for (int t = 0; t < 10; ++t) {
            float4 g = g4[t];
            float4 a = r0[t];
            float4 c = r1[t];
            acc0 = fmaf(g.x, a.x, acc0); acc0 = fmaf(g.y, a.y, acc0);
            acc0 = fmaf(g.z, a.z, acc0); acc0 = fmaf(g.w, a.w, acc0);
            acc1 = fmaf(g.x, c.x, acc1); acc1 = fmaf(g.y, c.y, acc1);
            acc1 = fmaf(g.z, c.z, acc1); acc1 = fmaf(g.w, c.w, acc1);
        }
    }
    float* zb = z + (size_t)bc * (H * OUTD);
    zb[(h0 + hg)      * OUTD + p0 + tp] = acc0;
    zb[(h0 + hg + 16) * OUTD + p0 + tp] = acc1;
}

// ---------------------------------------------------------------------------
// dsrow: out(16x16) = Avar(16x80) * Zblk(80x16) via 20x V_WMMA_F32_16X16X4.
// 4 waves/block, one 16-wide p tile each.
// ---------------------------------------------------------------------------
__global__ __launch_bounds__(128) void dsrow_kernel(const float* __restrict__ z,
                                                    const float* __restrict__ Avar,
                                                    float* __restrict__ out) {
    const int lane = threadIdx.x & 31;
    const int wave = threadIdx.x >> 5;
    const int bz   = blockIdx.z;
    const int p0   = (blockIdx.x * 4 + wave) * 16;   // 16 p tiles
    const int ot   = blockIdx.y;                     // 16 o tiles
    const int o0   = ot * 16;

    const int variant = (ot == 0) ? 0 : ((ot == 15) ? 2 : 1);
    const int base    = (ot == 0) ? 0 : ((ot == 15) ? 953 : 4 * o0 - 7);

    const float* A   = Avar + variant * 16 * AW;
    const float* zbc = z + (size_t)bz * (H * OUTD);

    const int ln   = lane & 15;
    const int koff = (lane < 16) ? 0 : 2;

    v8f acc;
    #pragma unroll
    for (int i = 0; i < 8; ++i) acc[i] = 0.f;

    for (int s = 0; s < 20; ++s) {
        const int k0 = 4 * s + koff;
        v2f a;
        a.x = A[ln * AW + k0];
        a.y = A[ln * AW + k0 + 1];
        int r0 = base + k0;
        int r1 = r0 + 1;
        r0 = (r0 > H - 1) ? (H - 1) : r0;   // rows past band carry zero weight
        r1 = (r1 > H - 1) ? (H - 1) : r1;
        v2f bm;
        bm.x = zbc[r0 * OUTD + p0 + ln];
        bm.y = zbc[r1 * OUTD + p0 + ln];
        acc = __builtin_amdgcn_wmma_f32_16x16x4_f32(
            false, a, false, bm, (short)0, acc, false, false);
    }

    const int mrow = (lane < 16) ? 0 : 8;
    #pragma unroll
    for (int v = 0; v < 8; ++v)
        out[((size_t)bz * OUTD + o0 + mrow + v) * OUTD + p0 + ln] = acc[v];
}

// ---------------------------------------------------------------------------
extern "C" void kernel_launch(void* const* d_in, const int* in_sizes, int n_in,
                              void* d_out, int out_size, void* d_ws, size_t ws_size,
                              hipStream_t stream) {
    const float* x = (const float*)d_in[0];   // (8,3,1024,1024)
    const float* k = (const float*)d_in[1];   // (8,1,21,21)
    float* out = (float*)d_out;               // (8,3,256,256)

    float* wsf  = (float*)d_ws;
    float* Avar = wsf;                        // 3*16*80   = 3840 floats
    float* gtab = wsf + 4096;                 // 8*5*21*40 = 33600 floats
    float* z    = wsf + 40960;                // 24*1024*256 floats (~25 MB)

    prep_avar<<<1, 256, 0, stream>>>(Avar);
    prep_gtab<<<dim3(64), 256, 0, stream>>>(k, gtab);
    convdsw_kernel<<<dim3(16, 32, BC), 256, 0, stream>>>(x, gtab, z);
    dsrow_kernel<<<dim3(4, 16, BC), 128, 0, stream>>>(z, Avar, out);
}